// MathematicallyCorrectGASM_66065186947099
// MI455X (gfx1250) — compile-verified
//
#include <hip/hip_runtime.h>
#include <hip/hip_bf16.h>
#include <math.h>

#define LR   0.01f
#define EPSF 1e-6f
#define N_PTS   8192
#define E_EDGES 262144
#define K_PAIRS 65536
#define ROW_WORDS (N_PTS / 32)   // 256 u32 words per adjacency row
#define RWAVES 4                 // waves per repulsion block

typedef float v2f __attribute__((ext_vector_type(2)));
typedef float v8f __attribute__((ext_vector_type(8)));

// ---------------------------------------------------------------- init
__global__ void init_kernel(const float* __restrict__ pos,
                            float4* __restrict__ xbuf,
                            unsigned int* __restrict__ bits /* 2*N*ROW_WORDS words */) {
    const size_t total_words = (size_t)2 * N_PTS * ROW_WORDS;
    const size_t stride = (size_t)gridDim.x * blockDim.x;
    for (size_t t = (size_t)blockIdx.x * blockDim.x + threadIdx.x; t < total_words; t += stride)
        bits[t] = 0u;
    for (size_t i = (size_t)blockIdx.x * blockDim.x + threadIdx.x; i < N_PTS; i += stride)
        xbuf[i] = make_float4(pos[3*i], pos[3*i+1], pos[3*i+2], 0.0f);
}

// ---------------------------------------------------------------- phase 1
__global__ void dist_kernel(const int* __restrict__ dist_idx,
                            const float* __restrict__ dist_target,
                            const float* __restrict__ pos,
                            float* __restrict__ xb /* float4-strided */) {
    int t = blockIdx.x * blockDim.x + threadIdx.x;
    if (t >= K_PAIRS) return;
    int i = dist_idx[2*t + 0];
    int j = dist_idx[2*t + 1];
    float vx = pos[3*i+0] - pos[3*j+0];
    float vy = pos[3*i+1] - pos[3*j+1];
    float vz = pos[3*i+2] - pos[3*j+2];
    float dist = sqrtf(vx*vx + vy*vy + vz*vz + 1e-12f);
    bool  valid = (dist > EPSF) && (i != j);
    float coef = valid ? (LR * 0.5f * (dist - dist_target[t]) / dist) : 0.0f;
    float cx = coef * vx, cy = coef * vy, cz = coef * vz;
    atomicAdd(&xb[4*i+0], -cx); atomicAdd(&xb[4*i+1], -cy); atomicAdd(&xb[4*i+2], -cz);
    atomicAdd(&xb[4*j+0],  cx); atomicAdd(&xb[4*j+1],  cy); atomicAdd(&xb[4*j+2],  cz);
}

// ---------------------------------------------------------------- phase 2
// All-pairs repulsion. 4 waves per 16-row tile split the j range; each wave
// computes 16x16 Gram tiles with V_WMMA_F32_16X16X4_F32; branchless mask via
// v_rsq; partials merged through LDS float atomics.
__global__ __launch_bounds__(32 * RWAVES)
void repulse_kernel(const float4* __restrict__ xbuf,
                    const float* __restrict__ min_dist,
                    float* __restrict__ xout) {
    __shared__ float acc[16][4];         // wsum, wx, wy, wz per row

    const int tid  = threadIdx.x;
    const int wave = tid >> 5;
    const int lane = tid & 31;
    const int half = lane >> 4;          // 0: K={0,1}, 1: K={2,3}
    const int l15  = lane & 15;
    const int i0   = blockIdx.x * 16;
    const float md = min_dist[0];

    if (tid < 64) ((float*)acc)[tid] = 0.0f;
    __syncthreads();

    // A tile: row m = l15; halves carry {x,y} / {z,0}
    float4 xi = xbuf[i0 + l15];
    v2f A;
    A[0] = half ? xi.z : xi.x;
    A[1] = half ? 0.0f : xi.y;

    float sqm[8];
#pragma unroll
    for (int k = 0; k < 8; ++k) {
        float4 xr = xbuf[i0 + k + 8*half];
        sqm[k] = xr.x*xr.x + xr.y*xr.y + xr.z*xr.z;
    }

    float wsum[8], wxa[8], wya[8], wza[8];
#pragma unroll
    for (int k = 0; k < 8; ++k) { wsum[k]=0.f; wxa[k]=0.f; wya[k]=0.f; wza[k]=0.f; }

    for (int j = wave * 16; j < N_PTS; j += 16 * RWAVES) {
        float4 xj = xbuf[j + l15];
        v2f B;
        B[0] = half ? xj.z : xj.x;
        B[1] = half ? 0.0f : xj.y;
        float sqn = xj.x*xj.x + xj.y*xj.y + xj.z*xj.z;

        v8f c = {0.f,0.f,0.f,0.f,0.f,0.f,0.f,0.f};
        c = __builtin_amdgcn_wmma_f32_16x16x4_f32(false, A, false, B,
                                                  (short)0, c, false, false);
#pragma unroll
        for (int k = 0; k < 8; ++k) {
            // element (m = k + 8*half, n = l15): g = x_m . x_n
            float d2   = fmaxf(fmaf(-2.0f, c[k], sqm[k] + sqn), 1e-12f);
            float rinv = __builtin_amdgcn_rsqf(d2);        // v_rsq_f32
            float dm   = d2 * rinv;                        // ~sqrt(d2)
            // (md-dm)/dm = md*rinv - 1 ; d2 > 1e-12 <=> reference dist > EPS
            bool  ok   = (dm < md) && (d2 > 1e-12f);
            float wv   = ok ? fmaf(md, rinv, -1.0f) : 0.0f;
            wsum[k] += wv;
            wxa[k]  += wv * xj.x;
            wya[k]  += wv * xj.y;
            wza[k]  += wv * xj.z;
        }
    }

    // reduce across the 16 lanes of each half (xor offsets stay in-half)
#pragma unroll
    for (int k = 0; k < 8; ++k) {
#pragma unroll
        for (int off = 1; off < 16; off <<= 1) {
            wsum[k] += __shfl_xor(wsum[k], off, 32);
            wxa[k]  += __shfl_xor(wxa[k],  off, 32);
            wya[k]  += __shfl_xor(wya[k],  off, 32);
            wza[k]  += __shfl_xor(wza[k],  off, 32);
        }
    }

    if (l15 == 0) {   // lanes 0/16 of each wave hold sums for rows 0-7 / 8-15
#pragma unroll
        for (int k = 0; k < 8; ++k) {
            int r = k + 8*half;
            atomicAdd(&acc[r][0], wsum[k]);   // ds_add_f32
            atomicAdd(&acc[r][1], wxa[k]);
            atomicAdd(&acc[r][2], wya[k]);
            atomicAdd(&acc[r][3], wza[k]);
        }
    }
    __syncthreads();

    if (tid < 16) {
        const float s = 0.5f * LR;
        int m = i0 + tid;
        float4 xr = xbuf[m];
        float ws = s * acc[tid][0];
        xout[3*m+0] = xr.x + xr.x * ws - s * acc[tid][1];
        xout[3*m+1] = xr.y + xr.y * ws - s * acc[tid][2];
        xout[3*m+2] = xr.z + xr.z * ws - s * acc[tid][3];
    }
}

// ---------------------------------------------------------------- phase 3
__global__ void edge_kernel(const int* __restrict__ e,
                            unsigned int* __restrict__ bitA,
                            unsigned int* __restrict__ bitB) {
    int t = blockIdx.x * blockDim.x + threadIdx.x;
    if (t >= E_EDGES) return;
    int a = e[t];           a = min(max(a, 0), N_PTS - 1);
    int b = e[E_EDGES + t]; b = min(max(b, 0), N_PTS - 1);
    atomicOr(&bitA[(size_t)a * ROW_WORDS + (b >> 5)], 1u << (b & 31));
    atomicOr(&bitB[(size_t)b * ROW_WORDS + (a >> 5)], 1u << (a & 31));
}

__global__ void degree_kernel(const unsigned int* __restrict__ bitA,
                              const unsigned int* __restrict__ bitB,
                              float* __restrict__ dinv) {
    int i = blockIdx.x * blockDim.x + threadIdx.x;
    if (i >= N_PTS) return;
    const uint4* rowA = (const uint4*)(bitA + (size_t)i * ROW_WORDS);
    const uint4* rowB = (const uint4*)(bitB + (size_t)i * ROW_WORDS);
    int deg = 0;
    for (int w = 0; w < ROW_WORDS / 4; ++w) {
        uint4 a = rowA[w], b = rowB[w];
        deg += __popc(a.x) + __popc(a.y) + __popc(a.z) + __popc(a.w);
        deg += __popc(b.x) + __popc(b.y) + __popc(b.z) + __popc(b.w);
    }
    dinv[i] = (deg == 0) ? 0.0f : rsqrtf((float)deg + EPSF);
}

// lap_i = x_i - dinv_i * sum_j adj_ij * dinv_j * x_j ; curvature = |lap_i|
__global__ __launch_bounds__(32)
void lap_kernel(const unsigned int* __restrict__ bitA,
                const unsigned int* __restrict__ bitB,
                const float* __restrict__ dinv,
                const float* __restrict__ xo,
                float* __restrict__ curv) {
    const int i = blockIdx.x;
    const int lane = threadIdx.x;
    const unsigned int* rowA = bitA + (size_t)i * ROW_WORDS;
    const unsigned int* rowB = bitB + (size_t)i * ROW_WORDS;
    float ax = 0.f, ay = 0.f, az = 0.f;
    for (int w = lane; w < ROW_WORDS; w += 32) {
        unsigned int wa = rowA[w];
        while (wa) {
            int b = __ffs(wa) - 1; wa &= wa - 1;
            int j = w * 32 + b;
            float dj = dinv[j];
            ax += dj * xo[3*j+0]; ay += dj * xo[3*j+1]; az += dj * xo[3*j+2];
        }
        unsigned int wb = rowB[w];
        while (wb) {
            int b = __ffs(wb) - 1; wb &= wb - 1;
            int j = w * 32 + b;
            float dj = dinv[j];
            ax += dj * xo[3*j+0]; ay += dj * xo[3*j+1]; az += dj * xo[3*j+2];
        }
    }
#pragma unroll
    for (int off = 1; off < 32; off <<= 1) {
        ax += __shfl_xor(ax, off, 32);
        ay += __shfl_xor(ay, off, 32);
        az += __shfl_xor(az, off, 32);
    }
    if (lane == 0) {
        float di = dinv[i];
        float lx = xo[3*i+0] - di * ax;
        float ly = xo[3*i+1] - di * ay;
        float lz = xo[3*i+2] - di * az;
        curv[i] = sqrtf(lx*lx + ly*ly + lz*lz + 1e-12f);
    }
}

// ---------------------------------------------------------------- launch
extern "C" void kernel_launch(void* const* d_in, const int* in_sizes, int n_in,
                              void* d_out, int out_size, void* d_ws, size_t ws_size,
                              hipStream_t stream) {
    const float* pos  = (const float*)d_in[0];
    const int*   edge = (const int*)d_in[1];
    const int*   didx = (const int*)d_in[2];
    const float* dtgt = (const float*)d_in[3];
    const float* mind = (const float*)d_in[4];
    float* out = (float*)d_out;

    char* ws = (char*)d_ws;
    float4*       xbuf = (float4*)ws;                                   // 128 KB
    unsigned int* bitA = (unsigned int*)(ws + 128 * 1024);              // 8 MB
    unsigned int* bitB = bitA + (size_t)N_PTS * ROW_WORDS;              // 8 MB
    float*        dinv = (float*)(bitB + (size_t)N_PTS * ROW_WORDS);    // 32 KB

    float* xout = out;                 // (N,3) flat
    float* curv = out + 3 * N_PTS;     // (N,)

    init_kernel<<<2048, 256, 0, stream>>>(pos, xbuf, bitA /* covers bitA+bitB */);
    dist_kernel<<<K_PAIRS / 256, 256, 0, stream>>>(didx, dtgt, pos, (float*)xbuf);
    repulse_kernel<<<N_PTS / 16, 32 * RWAVES, 0, stream>>>(xbuf, mind, xout);
    edge_kernel<<<E_EDGES / 256, 256, 0, stream>>>(edge, bitA, bitB);
    degree_kernel<<<N_PTS / 256, 256, 0, stream>>>(bitA, bitB, dinv);
    lap_kernel<<<N_PTS, 32, 0, stream>>>(bitA, bitB, dinv, xout, curv);
}